// QNet_44160853737570
// MI455X (gfx1250) — compile-verified
//
#include <hip/hip_runtime.h>
#include <hip/hip_bf16.h>
#include <math.h>

typedef __attribute__((ext_vector_type(2))) float v2f;
typedef __attribute__((ext_vector_type(8))) float v8f;

#define BB 2
#define LL 1024
#define VV 32000
#define HH 512
#define DD 512
#define MM (BB * LL)      // 2048 flattened rows
#define FF (DD + HH)      // 1024 fused width
#define NN 512            // output width of every GEMM in this net

// ---------------------------------------------------------------------------
// Embedding gather: emb[row, :] = W_emb[ids[row], :]   (float4 vectorized)
// ---------------------------------------------------------------------------
__global__ void gather_emb_kernel(const int* __restrict__ ids,
                                  const float* __restrict__ W_emb,
                                  float* __restrict__ emb) {
    const int per_row = HH / 4;  // 128 float4 per row
    int t = blockIdx.x * blockDim.x + threadIdx.x;
    if (t >= MM * per_row) return;
    int row = t / per_row;
    int c4  = t % per_row;
    int tok = ids[row];
    const float4* src = (const float4*)(W_emb + (size_t)tok * HH) + c4;
    float4* dst = (float4*)(emb + (size_t)row * HH) + c4;
    *dst = *src;
}

// ---------------------------------------------------------------------------
// Segment-end scan: position i is an END iff seg[i+1]==1 or i==L-1.
// Serial per batch (B=2, L=1024 — negligible).
// ---------------------------------------------------------------------------
__global__ void seg_ends_kernel(const int* __restrict__ seg,
                                int* __restrict__ end_idx,
                                int* __restrict__ nseg) {
    int b = blockIdx.x;
    if (threadIdx.x != 0) return;
    int cnt = 0;
    for (int i = 0; i < LL; ++i) {
        bool is_end = (i == LL - 1) || (seg[b * LL + i + 1] != 0);
        if (is_end) end_idx[b * LL + cnt++] = i;
    }
    nseg[b] = cnt;
}

// ---------------------------------------------------------------------------
// Build fused (M x 1024): row (b,i) = [dec_states[b,end_idx[i]], ctx_states[...]]
// for i < nseg[b], else zeros. (Bias is added by the following GEMM, matching
// the reference which applies the valid-mask BEFORE the mu/lv projections.)
// ---------------------------------------------------------------------------
__global__ void build_fused_kernel(const float* __restrict__ dec,
                                   const float* __restrict__ ctx,
                                   const int* __restrict__ end_idx,
                                   const int* __restrict__ nseg,
                                   float* __restrict__ fused) {
    const int per_row = FF / 4;  // 256 float4 per row
    int t = blockIdx.x * blockDim.x + threadIdx.x;
    if (t >= MM * per_row) return;
    int row = t / per_row;
    int c4  = t % per_row;
    int b = row / LL;
    int i = row % LL;
    float4 v = make_float4(0.f, 0.f, 0.f, 0.f);
    if (i < nseg[b]) {
        int j   = end_idx[b * LL + i];
        int col = c4 * 4;
        const float* src = (col < DD)
            ? dec + (size_t)(b * LL + j) * DD + col
            : ctx + (size_t)(b * LL + j) * HH + (col - DD);
        v = *(const float4*)src;
    }
    *(float4*)(fused + (size_t)row * FF + c4 * 4) = v;
}

// ---------------------------------------------------------------------------
// Split-K combine for the decoder projection: dec = p0 + p1 + b_proj
// Deterministic (fixed order, no atomics).
// ---------------------------------------------------------------------------
__global__ void combine_dec_kernel(const float* __restrict__ psum,
                                   const float* __restrict__ bias,
                                   float* __restrict__ out) {
    int t = blockIdx.x * blockDim.x + threadIdx.x;      // float4 index
    if (t >= MM * (NN / 4)) return;
    const float4* p0 = (const float4*)psum;
    const float4* p1 = p0 + (size_t)MM * (NN / 4);
    float4 a = p0[t];
    float4 b = p1[t];
    float4 bv = ((const float4*)bias)[t & (NN / 4 - 1)];
    float4 r;
    r.x = a.x + b.x + bv.x;
    r.y = a.y + b.y + bv.y;
    r.z = a.z + b.z + bv.z;
    r.w = a.w + b.w + bv.w;
    ((float4*)out)[t] = r;
}

// ---------------------------------------------------------------------------
// FP32 WMMA GEMM: C(M x 512) = act(A(M x K) @ B(K x 512) [+ bias])
// KCHUNK/LDA compile-time -> all loop loads are immediate-offset.
// One wave computes a 32x64 block: per k-step, 2 A fragments + 4 B fragments
// feed 8 WMMAs into 8 INDEPENDENT accumulator chains.
//   -> 1.25 VMEM loads per WMMA: VMEM pipe balanced against the WMMA pipe.
// gridDim.y = split-K index (chunk of KCHUNK); when !BIAS, raw partials are
// written to C + split*M*N for a deterministic combine pass.
// 8 waves/block cover the full 512-wide N of one 32-row strip -> the block
// shares its 32 A rows through L0/L2; W stays resident in the 192MB L2.
// All branches wave-uniform -> EXEC all-ones around WMMA (ISA requirement).
// ---------------------------------------------------------------------------
template <int KCHUNK, int LDA, bool GELU, bool BIAS>
__global__ void gemm32x64_f32_kernel(const float* __restrict__ A,
                                     const float* __restrict__ Bm,
                                     const float* __restrict__ bias,
                                     float* __restrict__ C) {
    int wave = threadIdx.x >> 5;
    int lane = threadIdx.x & 31;
    int tile = blockIdx.x * 8 + wave;   // 8 waves per block
    int tm   = tile >> 3;               // M-tile index (32 rows)
    int tng  = tile & 7;                // N-group index (64 cols)

    int m   = lane & 15;
    int khi = lane >> 4;                // which K-pair this half-wave owns
    int ks  = blockIdx.y * KCHUNK;      // split-K start

    const float* arow0 = A + (size_t)(tm * 32 + m) * LDA + 2 * khi + ks;
    const float* arow1 = arow0 + (size_t)16 * LDA;
    const float* bbase = Bm + (size_t)(ks + 2 * khi) * NN + tng * 64 + m;

    v8f acc[2][4] = {};
#pragma unroll 2
    for (int k0 = 0; k0 < KCHUNK; k0 += 4) {
        v2f a0 = *(const v2f*)(arow0 + k0);        // rows tm*32 .. +15
        v2f a1 = *(const v2f*)(arow1 + k0);        // rows tm*32+16 .. +31
        const float* bp = bbase + (size_t)k0 * NN;
#pragma unroll
        for (int c = 0; c < 4; ++c) {
            v2f b;
            b.x = bp[c * 16];                      // B[ks+k0+2khi  ][n+c*16]
            b.y = bp[c * 16 + NN];                 // B[ks+k0+2khi+1][n+c*16]
            acc[0][c] = __builtin_amdgcn_wmma_f32_16x16x4_f32(
                false, a0, false, b, (short)0, acc[0][c], false, false);
            acc[1][c] = __builtin_amdgcn_wmma_f32_16x16x4_f32(
                false, a1, false, b, (short)0, acc[1][c], false, false);
        }
    }

    float* Cp = BIAS ? C : (C + (size_t)blockIdx.y * MM * NN);
#pragma unroll
    for (int g = 0; g < 2; ++g) {
        int mbase = tm * 32 + g * 16 + 8 * khi;
#pragma unroll
        for (int c = 0; c < 4; ++c) {
            int n = tng * 64 + c * 16 + m;
            float bv = BIAS ? bias[n] : 0.0f;
#pragma unroll
            for (int r = 0; r < 8; ++r) {
                float x = acc[g][c][r] + bv;
                if (GELU) {
                    // jax.nn.gelu default (tanh approximation)
                    float x3 = x * x * x;
                    x = 0.5f * x * (1.0f + tanhf(0.7978845608028654f * (x + 0.044715f * x3)));
                }
                Cp[(size_t)(mbase + r) * NN + n] = x;
            }
        }
    }
}

// ---------------------------------------------------------------------------
// Host launcher
// ---------------------------------------------------------------------------
extern "C" void kernel_launch(void* const* d_in, const int* in_sizes, int n_in,
                              void* d_out, int out_size, void* d_ws, size_t ws_size,
                              hipStream_t stream) {
    const float* decoder_output = (const float*)d_in[0];   // (B,L,V)
    const int*   input_ids      = (const int*)  d_in[1];   // (B,L,1)
    const int*   seg            = (const int*)  d_in[2];   // (B,L,1)
    const float* W_emb          = (const float*)d_in[3];   // (V,H)
    const float* W_proj         = (const float*)d_in[4];   // (V,D)
    const float* b_proj         = (const float*)d_in[5];   // (D)
    const float* W_mix          = (const float*)d_in[6];   // (H,H)
    const float* b_mix          = (const float*)d_in[7];   // (H)
    const float* W_mu           = (const float*)d_in[8];   // (D+H,D)
    const float* b_mu           = (const float*)d_in[9];   // (D)
    const float* W_lv           = (const float*)d_in[10];  // (D+H,D)
    const float* b_lv           = (const float*)d_in[11];  // (D)

    float* out_mu = (float*)d_out;                         // (B,L,D)
    float* out_lv = out_mu + (size_t)MM * DD;              // (B,L,D)

    // Workspace layout (all regions 16B-aligned).
    // The split-K partial buffers (2 * M*D = 8MB) alias the `fused` region:
    // partials die at combine_dec, fused is built strictly afterwards.
    float* emb      = (float*)d_ws;                 // M*H
    float* ctxs     = emb  + (size_t)MM * HH;       // M*H
    float* decs     = ctxs + (size_t)MM * HH;       // M*D
    float* fused    = decs + (size_t)MM * DD;       // M*F  (aliases psum[2])
    float* psum     = fused;                        // 2 * M*D partials
    int*   end_idx  = (int*)(fused + (size_t)MM * FF);  // M
    int*   nseg     = end_idx + MM;                 // B

    const int gemm_blocks = (MM / 32) * (NN / 64) / 8;  // 8 waves/block, exact

    // 1) embedding gather
    {
        int total = MM * (HH / 4);
        gather_emb_kernel<<<(total + 255) / 256, 256, 0, stream>>>(input_ids, W_emb, emb);
    }
    // 2) ctx_states = gelu(emb @ W_mix + b_mix)            K = 512
    gemm32x64_f32_kernel<HH, HH, true, true>
        <<<dim3(gemm_blocks, 1), 256, 0, stream>>>(emb, W_mix, b_mix, ctxs);
    // 3) dec_states partials: decoder_output @ W_proj      K = 32000, split-K=2
    gemm32x64_f32_kernel<VV / 2, VV, false, false>
        <<<dim3(gemm_blocks, 2), 256, 0, stream>>>(decoder_output, W_proj, nullptr, psum);
    //    combine partials + b_proj -> decs (deterministic)
    {
        int total = MM * (NN / 4);
        combine_dec_kernel<<<(total + 255) / 256, 256, 0, stream>>>(psum, b_proj, decs);
    }
    // 4) segment-end positions per batch row
    seg_ends_kernel<<<BB, 32, 0, stream>>>(seg, end_idx, nseg);
    // 5) fused = [dec_rep, ctx_rep] with validity masking (overwrites psum region)
    {
        int total = MM * (FF / 4);
        build_fused_kernel<<<(total + 255) / 256, 256, 0, stream>>>(
            decs, ctxs, end_idx, nseg, fused);
    }
    // 6) mu_q / logvar_q projections                        K = 1024
    gemm32x64_f32_kernel<FF, FF, false, true>
        <<<dim3(gemm_blocks, 1), 256, 0, stream>>>(fused, W_mu, b_mu, out_mu);
    gemm32x64_f32_kernel<FF, FF, false, true>
        <<<dim3(gemm_blocks, 1), 256, 0, stream>>>(fused, W_lv, b_lv, out_lv);
}